// NMU_11175504904631
// MI455X (gfx1250) — compile-verified
//
#include <hip/hip_runtime.h>
#include <stdint.h>

// NMU forward: y[b,o] = prod_d fma( clip(M[d,o],0,1), x[b,d] - 1, 1.0 )
// B=16384, D=256, O=32.
// Mapping: block = 256 threads (8 wave32), 32 rows/block.
//   wave -> 4 rows; lane = (g = row-group 0..3, j = o-quad 0..7)
//   each lane: 4 outputs (two packed float2 accumulators) -> v_pk_fma_f32/v_pk_mul_f32
// CDNA5 paths: global_load_async_to_lds_b128 (ASYNCcnt) + s_wait_asynccnt,
//              global_prefetch_b8, nontemporal b128 stores.

#define DDIM     256
#define ODIM     32
#define ROWS     32      // rows (b) per block
#define TPB      256     // 8 wave32s
#define XSTRIDE  260     // D + 4 pad: x broadcast reads for the 4 row-groups hit disjoint banks

typedef __attribute__((ext_vector_type(4))) float v4f;
typedef __attribute__((ext_vector_type(2))) float v2f;

__global__ __launch_bounds__(TPB) void nmu_fwd_kernel(const float* __restrict__ x,
                                                      const float* __restrict__ M,
                                                      float* __restrict__ out,
                                                      int rowsTotal) {
    __shared__ float sM[DDIM * ODIM];      // 32 KB: clip(M,0,1), row-major (as in global)
    __shared__ float sX[ROWS * XSTRIDE];   // 32.5 KB: x - 1, padded row stride

    const int tid   = threadIdx.x;
    const int wave  = tid >> 5;
    const int lane  = tid & 31;
    const int g     = lane >> 3;           // row group within wave: 0..3
    const int j     = lane & 7;            // o-quad within row:     0..7
    const int obase = j * 4;
    const int rloc  = wave * 4 + g;        // local row 0..31
    const int b0    = blockIdx.x * ROWS;

    // gfx1250 prefetch path for the shared M matrix
    __builtin_prefetch(M, 0, 1);

    // ---- Async-DMA 32 rows of x (32 KB, contiguous in global) into padded LDS.
    //      2048 chunks of 16 B; 64 chunks per row; per-lane scatter handles the pad.
    {
        uint64_t gbase = (uint64_t)(uintptr_t)(x + (size_t)b0 * DDIM);
        uint32_t lbase = (uint32_t)(uintptr_t)(&sX[0]);  // lo32 of generic = LDS offset
#pragma unroll
        for (int p = 0; p < 8; ++p) {
            int c      = p * TPB + tid;                  // chunk id 0..2047
            int row    = c >> 6;
            int col16  = c & 63;
            uint32_t ldsoff = lbase + (uint32_t)(row * (XSTRIDE * 4) + col16 * 16);
            uint64_t ga     = gbase + (uint64_t)c * 16;
            asm volatile("global_load_async_to_lds_b128 %0, %1, off"
                         :: "v"(ldsoff), "v"(ga)
                         : "memory");
        }
    }

    // ---- Stage clip(M,0,1) (overlaps with the async copy); coalesced, no transpose.
    for (int i = tid; i < DDIM * ODIM; i += TPB) {
        float m = M[i];
        sM[i] = fminf(fmaxf(m, 0.0f), 1.0f);
    }

    // Each wave waits for its own async ops; barrier publishes everything.
    asm volatile("s_wait_asynccnt 0" ::: "memory");
    __syncthreads();

    // ---- Convert x -> (x - 1) in place: thread owns row tid/8, 32 cols at (tid%8)*32.
    {
        v4f* px = (v4f*)&sX[(tid >> 3) * XSTRIDE + (tid & 7) * 32];
#pragma unroll
        for (int q = 0; q < 8; ++q) px[q] = px[q] - 1.0f;
    }
    __syncthreads();

    // ---- Main product. Per d: one v4f M-row slice (8 unique 16B addrs/wave, bank-clean),
    //      x scalar broadcast per row-group, 4 packed FMA/MUL per lane covering 4 outputs.
    const float* xrow = &sX[rloc * XSTRIDE];
    const float* mcol = &sM[obase];

    v2f acc01 = {1.0f, 1.0f};
    v2f acc23 = {1.0f, 1.0f};
    const v2f one2 = {1.0f, 1.0f};

#pragma unroll 4
    for (int d = 0; d < DDIM; d += 4) {
        v4f xv = *(const v4f*)(xrow + d);   // same addr within a row group -> LDS broadcast
#pragma unroll
        for (int k = 0; k < 4; ++k) {
            v4f mv = *(const v4f*)(mcol + (d + k) * ODIM);
            float xs = (k == 0) ? xv.x : (k == 1) ? xv.y : (k == 2) ? xv.z : xv.w;
            v2f xk = {xs, xs};
            acc01 *= __builtin_elementwise_fma((v2f){mv.x, mv.y}, xk, one2);
            acc23 *= __builtin_elementwise_fma((v2f){mv.z, mv.w}, xk, one2);
        }
    }

    const int b = b0 + rloc;
    if (b < rowsTotal) {
        v4f r = {acc01.x, acc01.y, acc23.x, acc23.y};
        __builtin_nontemporal_store(r, (v4f*)(out + (size_t)b * ODIM + obase));
    }
}

extern "C" void kernel_launch(void* const* d_in, const int* in_sizes, int n_in,
                              void* d_out, int out_size, void* d_ws, size_t ws_size,
                              hipStream_t stream) {
    (void)n_in; (void)out_size; (void)d_ws; (void)ws_size;
    const float* x = (const float*)d_in[0];   // [B, D] fp32
    const float* M = (const float*)d_in[1];   // [D, O] fp32
    float* out     = (float*)d_out;           // [B, O] fp32

    const int Brows = in_sizes[0] / DDIM;     // 16384
    dim3 grid((Brows + ROWS - 1) / ROWS);     // 512 blocks
    dim3 block(TPB);                          // 8 wave32s
    nmu_fwd_kernel<<<grid, block, 0, stream>>>(x, M, out, Brows);
}